// Quantizer_57939108823771
// MI455X (gfx1250) — compile-verified
//
#include <hip/hip_runtime.h>
#include <hip/hip_bf16.h>

typedef __attribute__((ext_vector_type(2))) float v2f;
typedef __attribute__((ext_vector_type(8))) float v8f;

// Problem shape (fixed by reference): inputs [16,64,64,64] f32, codebook [1024,64] f32
#define NB 16
#define NC 64           // channels == D
#define NHW 4096        // H*W
#define NROWS 65536     // NB*NHW
#define NK 1024
#define CHW 262144      // NC*NHW

// d_ws layout (floats): cbT[64*1024] | c2[1024] | counts[1024] | lsum[1]
#define WS_CBT    0
#define WS_C2     (NC * NK)
#define WS_COUNTS (WS_C2 + NK)
#define WS_LSUM   (WS_COUNTS + NK)

// ---------------------------------------------------------------------------
// Prep: transpose codebook to cbT[k][code] (coalesced B loads in main kernel),
// codebook squared norms, zero histogram + loss accumulator.
// ---------------------------------------------------------------------------
__global__ __launch_bounds__(256) void vq_prep(const float* __restrict__ cb,
                                               float* __restrict__ cbT,
                                               float* __restrict__ c2,
                                               float* __restrict__ counts,
                                               float* __restrict__ lsum) {
    int tid = blockIdx.x * 256 + threadIdx.x;   // 4 blocks -> tid in [0,1024)
    if (tid < NK) {
        const float* r = cb + tid * NC;
        float s = 0.0f;
        #pragma unroll 8
        for (int k = 0; k < NC; ++k) {
            float v = r[k];
            s += v * v;
            cbT[k * NK + tid] = v;              // transpose: cbT[k][code]
        }
        c2[tid] = s;
        counts[tid] = 0.0f;
    }
    if (tid == 0) *lsum = 0.0f;
}

// ---------------------------------------------------------------------------
// Main: per-wave 32 rows (two 16-row A tiles sharing B); distances via
// v_wmma_f32_16x16x4_f32; argmin; gather + loss + histogram + writeout.
// 128 threads/block = 4 waves; 512 blocks.
// ---------------------------------------------------------------------------
__global__ __launch_bounds__(128) void vq_main(const float* __restrict__ x,
                                               const float* __restrict__ cb,
                                               const float* __restrict__ cbT,
                                               const float* __restrict__ c2,
                                               float* __restrict__ counts,
                                               float* __restrict__ lsum,
                                               float* __restrict__ qout) {
    __shared__ float xs[4][32][68];   // stride 68 -> conflict-free banks
    __shared__ int   widx[4][32];

    const int lane  = threadIdx.x & 31;
    const int wave  = threadIdx.x >> 5;
    const int row_l = lane & 15;      // row within 16-row tile
    const int chalf = lane >> 4;      // K-half this lane supplies (A/B layout)

    const int n0 = blockIdx.x * 128 + wave * 32;
    const int b  = n0 >> 12;          // n0 / NHW (32 | 4096, never crosses b)
    const int hw = n0 & (NHW - 1);

    // ---- load 32x64 x tile, coalesced (16 consecutive addresses per channel)
    const float* xbase = x + (size_t)b * CHW + hw;   // + c*NHW + row
    #pragma unroll
    for (int rr = 0; rr < 2; ++rr) {
        #pragma unroll 4
        for (int i = 0; i < 32; ++i) {
            int c = 2 * i + chalf;
            xs[wave][rr * 16 + row_l][c] = xbase[c * NHW + rr * 16 + row_l];
        }
    }
    asm volatile("s_wait_dscnt 0" ::: "memory");     // intra-wave LDS visibility

    float best0[8], best1[8];
    int   bidx0[8], bidx1[8];
    #pragma unroll
    for (int r = 0; r < 8; ++r) {
        best0[r] = 3.4e38f; bidx0[r] = 0;
        best1[r] = 3.4e38f; bidx1[r] = 0;
    }

    const float* arow0 = &xs[wave][row_l][0];        // rows 0..15
    const float* arow1 = &xs[wave][16 + row_l][0];   // rows 16..31

    // B base: cbT[k][code]; lane supplies codes t*16+row_l at k = 4kc+2chalf(+1)
    const float* brow = cbT + chalf * (2 * NK) + row_l;

    for (int t = 0; t < NK / 16; ++t) {
        // prefetch next tile's B lines (global_prefetch_b8)
        if (t + 1 < NK / 16) {
            __builtin_prefetch(brow + (t + 1) * 16, 0, 0);
            __builtin_prefetch(brow + (t + 1) * 16 + NK, 0, 0);
        }
        v8f acc0 = {};
        v8f acc1 = {};
        const float* bt = brow + t * 16;
        #pragma unroll
        for (int kc = 0; kc < 16; ++kc) {
            // two coalesced b32 loads (16 consecutive codes per half-wave)
            v2f bb = { bt[kc * 4 * NK], bt[kc * 4 * NK + NK] };
            v2f a0 = *(const v2f*)(arow0 + kc * 4 + 2 * chalf);
            v2f a1 = *(const v2f*)(arow1 + kc * 4 + 2 * chalf);
            acc0 = __builtin_amdgcn_wmma_f32_16x16x4_f32(
                false, a0, false, bb, (short)0, acc0, false, false);
            acc1 = __builtin_amdgcn_wmma_f32_16x16x4_f32(
                false, a1, false, bb, (short)0, acc1, false, false);
        }
        // distance key: ||c||^2 - 2*x.c  (||x||^2 constant per row -> dropped)
        float c2v  = c2[t * 16 + row_l];
        int   code = t * 16 + row_l;
        #pragma unroll
        for (int r = 0; r < 8; ++r) {
            float d0 = __builtin_fmaf(-2.0f, acc0[r], c2v);
            float d1 = __builtin_fmaf(-2.0f, acc1[r], c2v);
            if (d0 < best0[r]) { best0[r] = d0; bidx0[r] = code; }
            if (d1 < best1[r]) { best1[r] = d1; bidx1[r] = code; }
        }
    }

    // ---- cross-lane argmin within each 16-lane half (wave32 shuffles)
    #pragma unroll
    for (int r = 0; r < 8; ++r) {
        float v0 = best0[r]; int i0 = bidx0[r];
        float v1 = best1[r]; int i1 = bidx1[r];
        #pragma unroll
        for (int m = 1; m < 16; m <<= 1) {
            float ov0 = __shfl_xor(v0, m, 32); int oi0 = __shfl_xor(i0, m, 32);
            float ov1 = __shfl_xor(v1, m, 32); int oi1 = __shfl_xor(i1, m, 32);
            if (ov0 < v0 || (ov0 == v0 && oi0 < i0)) { v0 = ov0; i0 = oi0; }
            if (ov1 < v1 || (ov1 == v1 && oi1 < i1)) { v1 = ov1; i1 = oi1; }
        }
        bidx0[r] = i0; bidx1[r] = i1;
    }
    // lanes 0 / 16 publish the 32 per-row winners (C layout: VGPR r -> M=r, r+8)
    if (row_l == 0) {
        #pragma unroll
        for (int r = 0; r < 8; ++r) {
            widx[wave][r + 8 * chalf]      = bidx0[r];
            widx[wave][16 + r + 8 * chalf] = bidx1[r];
        }
    }
    asm volatile("s_wait_dscnt 0" ::: "memory");

    // ---- gather q, accumulate ||q - x||^2, histogram, write q to [B,C,H,W]
    float acc2 = 0.0f;
    #pragma unroll
    for (int rr = 0; rr < 2; ++rr) {
        const int row   = rr * 16 + row_l;
        const int myidx = widx[wave][row];
        if (lane < 16) atomicAdd(&counts[myidx], 1.0f);
        const float* cbr = cb + myidx * NC;
        float*       qb  = qout + (size_t)b * CHW + hw + row;
        #pragma unroll 4
        for (int ci = 0; ci < 32; ++ci) {
            int c = 2 * ci + chalf;
            float qv = cbr[c];
            float xv = xs[wave][row][c];
            float dd = qv - xv;
            acc2 = __builtin_fmaf(dd, dd, acc2);
            qb[c * NHW] = qv;      // 16 consecutive lanes -> 64B stores
        }
    }
    #pragma unroll
    for (int m = 1; m < 32; m <<= 1) acc2 += __shfl_xor(acc2, m, 32);
    if (lane == 0) atomicAdd(lsum, acc2);
}

// ---------------------------------------------------------------------------
// Finalize: loss scalar + perplexity
// ---------------------------------------------------------------------------
__global__ __launch_bounds__(1024) void vq_fin(const float* __restrict__ counts,
                                               const float* __restrict__ lsum,
                                               const float* __restrict__ beta,
                                               float* __restrict__ out_loss,
                                               float* __restrict__ out_ppl) {
    __shared__ float red[32];
    int tid = threadIdx.x;
    float e = counts[tid] * (1.0f / (float)NROWS);
    float p = e * __logf(e + 1e-10f);
    #pragma unroll
    for (int m = 1; m < 32; m <<= 1) p += __shfl_xor(p, m, 32);
    if ((tid & 31) == 0) red[tid >> 5] = p;
    __syncthreads();
    if (tid < 32) {
        float v = red[tid];
        #pragma unroll
        for (int m = 1; m < 32; m <<= 1) v += __shfl_xor(v, m, 32);
        if (tid == 0) {
            *out_ppl = __expf(-v);
            float mean = *lsum * (1.0f / (float)(NROWS * NC));
            // beta*mean((q-sg x)^2) + mean((sg q - x)^2) == (1+beta)*mean, *KLD_SCALE
            *out_loss = (1.0f + *beta) * 10.0f * mean;
        }
    }
}

// ---------------------------------------------------------------------------
extern "C" void kernel_launch(void* const* d_in, const int* in_sizes, int n_in,
                              void* d_out, int out_size, void* d_ws, size_t ws_size,
                              hipStream_t stream) {
    const float* x    = (const float*)d_in[0];   // [16,64,64,64]
    const float* cb   = (const float*)d_in[1];   // [1024,64]
    const float* beta = (const float*)d_in[2];   // scalar

    float* out = (float*)d_out;                  // [loss | quantized(4194304) | ppl]
    float* ws     = (float*)d_ws;
    float* cbT    = ws + WS_CBT;                 // [64*1024] transposed codebook
    float* c2     = ws + WS_C2;                  // [1024]
    float* counts = ws + WS_COUNTS;              // [1024]
    float* acc    = ws + WS_LSUM;                // [1]

    vq_prep<<<4, 256, 0, stream>>>(cb, cbT, c2, counts, acc);
    vq_main<<<NROWS / 128, 128, 0, stream>>>(x, cb, cbT, c2, counts, acc, out + 1);
    vq_fin<<<1, 1024, 0, stream>>>(counts, acc, beta, out, out + 1 + (size_t)NROWS * NC);
}